// GraphAttentionLayer_58420145160346
// MI455X (gfx1250) — compile-verified
//
#include <hip/hip_runtime.h>

typedef float v2f __attribute__((ext_vector_type(2)));
typedef float v4f __attribute__((ext_vector_type(4)));
typedef float v8f __attribute__((ext_vector_type(8)));

#define BB   8
#define NN   2048
#define FIN  256
#define FOUT 256
#define HH   8
#define DD   32
#define QSZ  (BB*HH*NN*DD)      /* 4194304 floats per Q/K/V buffer */
#define OUT0 (BB*NN*FOUT)       /* 4194304 floats, first output   */
#define SSTR 2052               /* padded LDS row stride, %4==0 -> 16B aligned rows */
#define TSTR 260                /* padded LDS row stride (proj), %4==0 */

static __device__ __forceinline__ v8f wmma_f32(v2f a, v2f b, v8f c) {
  // D(16x16,f32) = A(16x4,f32) * B(4x16,f32) + C
  return __builtin_amdgcn_wmma_f32_16x16x4_f32(false, a, false, b, (short)0, c,
                                               false, false);
}

// ---------------------------------------------------------------------------
// Kernel 1: QKV projection.  One wave per 16x16 output tile.
// grid = (M/16, 256/16, 3), block = 32.
// Writes Q/K/V in [B][H][N][D] layout (D contiguous).
// ---------------------------------------------------------------------------
__global__ void qkv_kernel(const float* __restrict__ x,
                           const float* __restrict__ Wq, const float* __restrict__ bq,
                           const float* __restrict__ Wk, const float* __restrict__ bk,
                           const float* __restrict__ Wv, const float* __restrict__ bv,
                           float* __restrict__ ws) {
  const int lane = threadIdx.x;
  const int r15  = lane & 15;
  const int half = lane >> 4;
  const int ko   = half * 2;
  const int m0   = blockIdx.x * 16;
  const int n0   = blockIdx.y * 16;
  const int z    = blockIdx.z;

  const float* W    = (z == 0) ? Wq : (z == 1) ? Wk : Wv;
  const float* bias = (z == 0) ? bq : (z == 1) ? bk : bv;
  float*       dst  = ws + (size_t)z * QSZ;

  const float* arow = x + (size_t)(m0 + r15) * FIN;
  v8f acc = {};
#pragma unroll 8
  for (int k = 0; k < FIN; k += 4) {
    v2f a = *(const v2f*)(arow + k + ko);       // A[row][k+ko], A[row][k+ko+1]
    v2f bf;
    bf.x = W[(k + ko) * FOUT + n0 + r15];       // B[k+ko][col]
    bf.y = W[(k + ko + 1) * FOUT + n0 + r15];   // B[k+ko+1][col]
    acc = wmma_f32(a, bf, acc);
  }

  const int   c   = n0 + r15;
  const float bb  = bias[c];
  const int   hId = c >> 5, d = c & 31;
  const int   bId = m0 >> 11;                   // batch index (tile never spans)
  const int   nb  = (m0 & (NN - 1)) + half * 8;
#pragma unroll
  for (int i = 0; i < 8; ++i) {
    int n = nb + i;                             // VGPR i -> row i / i+8
    dst[((size_t)(bId * HH + hId) * NN + n) * DD + d] = acc[i] + bb;
  }
}

// ---------------------------------------------------------------------------
// Kernel 2: masked attention + softmax + attn@V + head-mean of attention.
// grid = (N/16, B), block = 128 (4 waves), dynamic LDS ~265 KB:
//   S  [16][2052]  score / prob row-block for current head
//   AS [16][2052]  attention accumulated over heads
//   RED[4][512]    cross-wave reduction of 16x32 outputs
//   R2 [256]       softmax row max / row sum scratch
// All bulk LDS / global traffic is 128-bit vectorized.
// ---------------------------------------------------------------------------
__global__ void attn_kernel(const float* __restrict__ ws,
                            const int* __restrict__ adj,
                            float* __restrict__ ho,
                            float* __restrict__ amean) {
  extern __shared__ float smem[];
  float* S   = smem;
  float* AS  = S + 16 * SSTR;
  float* RED = AS + 16 * SSTR;
  float* R2  = RED + 4 * 512;

  const int tid  = threadIdx.x;
  const int wv   = tid >> 5;
  const int lane = tid & 31;
  const int r15  = lane & 15;
  const int half = lane >> 4;
  const int ko   = half * 2;
  const int b    = blockIdx.y;
  const int q0   = blockIdx.x * 16;
  const int r    = tid & 15;        // softmax row
  const int ch   = tid >> 4;        // softmax column chunk (8 x 256)
  const float scale = 0.17677669529663687f;   // 1/sqrt(32)

  const float* Q = ws;
  const float* K = ws + QSZ;
  const float* V = ws + 2 * (size_t)QSZ;

  {
    v4f z = {};
    v4f* AS4 = (v4f*)AS;
    for (int e = tid; e < 16 * SSTR / 4; e += 128) AS4[e] = z;
  }
  __syncthreads();

  for (int h = 0; h < HH; ++h) {
    const float* Qb = Q + (size_t)((b * HH + h) * NN) * DD;
    const float* Kb = K + (size_t)((b * HH + h) * NN) * DD;
    const float* Vb = V + (size_t)((b * HH + h) * NN) * DD;

    // ---- phase 1: S = (Q @ K^T) * scale, masked by adj ----
    const float* qrow = Qb + (size_t)(q0 + r15) * DD;
    for (int kt = wv; kt < NN / 16; kt += 4) {
      const int k0 = kt * 16;
      const float* kcol = Kb + (size_t)(k0 + r15) * DD;
      v8f acc = {};
#pragma unroll
      for (int kd = 0; kd < DD; kd += 4) {
        v2f a  = *(const v2f*)(qrow + kd + ko);   // Q[q][kd+ko..+1]
        v2f bf = *(const v2f*)(kcol + kd + ko);   // K[k][kd+ko..+1] (B = K^T)
        acc = wmma_f32(a, bf, acc);
      }
#pragma unroll
      for (int i = 0; i < 8; ++i) {
        const int qr = i + 8 * half;
        const int gk = k0 + r15;
        const int m  = adj[((size_t)(b * NN + q0 + qr)) * NN + gk];
        S[qr * SSTR + gk] = (m == 0) ? -__builtin_inff() : acc[i] * scale;
      }
    }
    __syncthreads();

    // ---- phase 2: row softmax over 2048 (v4 LDS), accumulate into AS ----
    {
      v4f* Srow4 = (v4f*)(S + r * SSTR + ch * 256);
      v4f* Arow4 = (v4f*)(AS + r * SSTR + ch * 256);
      float mx = -__builtin_inff();
      for (int j = 0; j < 64; ++j) {
        v4f t = Srow4[j];
        mx = fmaxf(mx, fmaxf(fmaxf(t.x, t.y), fmaxf(t.z, t.w)));
      }
      R2[ch * 16 + r] = mx;
      __syncthreads();
      float rowmax = -__builtin_inff();
#pragma unroll
      for (int c = 0; c < 8; ++c) rowmax = fmaxf(rowmax, R2[c * 16 + r]);
      float lsum = 0.0f;
      for (int j = 0; j < 64; ++j) {
        v4f t = Srow4[j];
        v4f e;
        e.x = __expf(t.x - rowmax);
        e.y = __expf(t.y - rowmax);
        e.z = __expf(t.z - rowmax);
        e.w = __expf(t.w - rowmax);
        Srow4[j] = e;
        lsum += (e.x + e.y) + (e.z + e.w);
      }
      R2[128 + ch * 16 + r] = lsum;
      __syncthreads();
      float rowsum = 0.0f;
#pragma unroll
      for (int c = 0; c < 8; ++c) rowsum += R2[128 + c * 16 + r];
      const float inv = 1.0f / rowsum;
      for (int j = 0; j < 64; ++j) {
        v4f p = Srow4[j] * inv;
        Srow4[j] = p;
        Arow4[j] += p;
      }
    }
    __syncthreads();

    // ---- phase 3: O(16x32) = P(16x2048) @ V(2048x32), split over 4 waves ----
    v8f o0 = {}, o1 = {};
    for (int kt = wv; kt < NN / 16; kt += 4) {
      const int k0 = kt * 16;
#pragma unroll
      for (int kk = 0; kk < 16; kk += 4) {
        v2f a = *(const v2f*)(S + r15 * SSTR + k0 + kk + ko);  // P[q][k..k+1] (LDS b64)
        const float* v0 = Vb + (size_t)(k0 + kk + ko) * DD;
        const float* v1 = v0 + DD;
        v2f b0, b1;
        b0.x = v0[r15];      b0.y = v1[r15];
        b1.x = v0[16 + r15]; b1.y = v1[16 + r15];
        o0 = wmma_f32(a, b0, o0);
        o1 = wmma_f32(a, b1, o1);
      }
    }
#pragma unroll
    for (int i = 0; i < 8; ++i) {
      const int qr = i + 8 * half;
      RED[wv * 512 + qr * 32 + r15]      = o0[i];
      RED[wv * 512 + qr * 32 + 16 + r15] = o1[i];
    }
    __syncthreads();
    {
      // 128 threads x one v4 each covers the 16x32 tile
      const int qr = tid >> 3;
      const int c4 = (tid & 7) * 4;
      v4f s = *(const v4f*)(RED + qr * 32 + c4);
      s += *(const v4f*)(RED + 512 + qr * 32 + c4);
      s += *(const v4f*)(RED + 1024 + qr * 32 + c4);
      s += *(const v4f*)(RED + 1536 + qr * 32 + c4);
      *(v4f*)(ho + ((size_t)(b * NN + q0 + qr)) * (HH * DD) + h * DD + c4) = s;
    }
    __syncthreads();
  }

  // ---- write attn.mean(axis=heads) once, b128 stores ----
  float* am = amean + ((size_t)(b * NN + q0)) * NN;
  for (int e = tid; e < 16 * (NN / 4); e += 128) {
    const int rr = e >> 9;               // 512 quads per row
    const int c4 = (e & 511) * 4;
    v4f t = *(const v4f*)(AS + rr * SSTR + c4);
    *(v4f*)(am + (size_t)rr * NN + c4) = t * (1.0f / HH);
  }
}

// ---------------------------------------------------------------------------
// Kernel 3: out = LN(ho @ Wo + bo) * gamma + beta.
// grid = M/16, block = 128 (4 waves); 16x256 tile staged in LDS for LN.
// ---------------------------------------------------------------------------
__global__ void proj_ln_kernel(const float* __restrict__ ho,
                               const float* __restrict__ Wo,
                               const float* __restrict__ bo,
                               const float* __restrict__ gamma,
                               const float* __restrict__ beta,
                               float* __restrict__ out) {
  __shared__ float T[16 * TSTR];
  __shared__ float R2[256];

  const int tid  = threadIdx.x;
  const int wv   = tid >> 5;
  const int lane = tid & 31;
  const int r15  = lane & 15;
  const int half = lane >> 4;
  const int ko   = half * 2;
  const int m0   = blockIdx.x * 16;

  const float* arow = ho + (size_t)(m0 + r15) * (HH * DD);
  for (int t = 0; t < 4; ++t) {
    const int n0 = (wv + 4 * t) * 16;
    v8f acc = {};
#pragma unroll 8
    for (int k = 0; k < HH * DD; k += 4) {
      v2f a = *(const v2f*)(arow + k + ko);
      v2f bf;
      bf.x = Wo[(k + ko) * FOUT + n0 + r15];
      bf.y = Wo[(k + ko + 1) * FOUT + n0 + r15];
      acc = wmma_f32(a, bf, acc);
    }
    const float bb = bo[n0 + r15];
#pragma unroll
    for (int i = 0; i < 8; ++i)
      T[(i + 8 * half) * TSTR + n0 + r15] = acc[i] + bb;
  }
  __syncthreads();

  const int r  = tid & 15;
  const int ch = tid >> 4;                 // 8 chunks of 32 cols
  const v4f* trow4 = (const v4f*)(T + r * TSTR + ch * 32);
  float s = 0.0f, sq = 0.0f;
#pragma unroll
  for (int j = 0; j < 8; ++j) {
    v4f t = trow4[j];
    s  += (t.x + t.y) + (t.z + t.w);
    sq += (t.x * t.x + t.y * t.y) + (t.z * t.z + t.w * t.w);
  }
  R2[ch * 16 + r] = s;
  R2[128 + ch * 16 + r] = sq;
  __syncthreads();
  float mean = 0.0f, ms = 0.0f;
#pragma unroll
  for (int c = 0; c < 8; ++c) { mean += R2[c * 16 + r]; ms += R2[128 + c * 16 + r]; }
  mean *= (1.0f / FOUT);
  ms   *= (1.0f / FOUT);
  const float rstd = rsqrtf(ms - mean * mean + 1e-5f);

  float* orow = out + (size_t)(m0 + r) * FOUT;
#pragma unroll
  for (int j = 0; j < 8; ++j) {
    const int c = ch * 32 + j * 4;
    v4f t = trow4[j];
    v4f g = *(const v4f*)(gamma + c);
    v4f be = *(const v4f*)(beta + c);
    v4f o = (t - mean) * rstd;
    o = o * g + be;
    *(v4f*)(orow + c) = o;
  }
}

// ---------------------------------------------------------------------------
extern "C" void kernel_launch(void* const* d_in, const int* in_sizes, int n_in,
                              void* d_out, int out_size, void* d_ws, size_t ws_size,
                              hipStream_t stream) {
  (void)in_sizes; (void)n_in; (void)out_size; (void)ws_size;
  const float* x     = (const float*)d_in[0];
  const int*   adj   = (const int*)d_in[1];
  const float* Wq    = (const float*)d_in[2];
  const float* bq    = (const float*)d_in[3];
  const float* Wk    = (const float*)d_in[4];
  const float* bk    = (const float*)d_in[5];
  const float* Wv    = (const float*)d_in[6];
  const float* bv    = (const float*)d_in[7];
  const float* Wo    = (const float*)d_in[8];
  const float* bo    = (const float*)d_in[9];
  const float* gamma = (const float*)d_in[10];
  const float* beta  = (const float*)d_in[11];
  float* out = (float*)d_out;
  float* ws  = (float*)d_ws;   // Q | K | V | HO  = 4 * 16 MB

  qkv_kernel<<<dim3(BB * NN / 16, FOUT / 16, 3), 32, 0, stream>>>(
      x, Wq, bq, Wk, bk, Wv, bv, ws);

  const size_t smem = (size_t)(2 * 16 * SSTR + 4 * 512 + 256) * sizeof(float);
  attn_kernel<<<dim3(NN / 16, BB), 128, smem, stream>>>(
      ws, adj, ws + 3 * (size_t)QSZ, out + OUT0);

  proj_ln_kernel<<<dim3(BB * NN / 16), 128, 0, stream>>>(
      ws + 3 * (size_t)QSZ, Wo, bo, gamma, beta, out);
}